// Diagnosis_1640677507708
// MI455X (gfx1250) — compile-verified
//
#include <hip/hip_runtime.h>
#include <hip/hip_bf16.h>

// ---------------------------------------------------------------------------
// Problem sizes (from the reference)
// ---------------------------------------------------------------------------
#define N_STU   10000
#define N_ITEM  50000
#define N_CONC  2048
#define EDGES   250000
#define KCONC   4
#define EMB     64
#define CNUM    128

typedef __attribute__((ext_vector_type(2))) float v2f;
typedef __attribute__((ext_vector_type(8))) float v8f;

// ---------------------------------------------------------------------------
// Kernel 1: Out[m][n] = sum_{k<64} A[m][k] * W[n][k]
//   A: M x 64 (row-major, stride 64)
//   W: rows of stride ldw (=128); pointer pre-offset to select the 64-col slice
//   Out: M x 128 (row-major, stride 128)
// One wave (32 lanes) computes one 16x16 tile with 16 chained
// V_WMMA_F32_16X16X4_F32 ops (fp32 in/out -> exact reference numerics).
// Block = (32,8): wave y owns column tile n0 = 16*y. grid.x = M/16.
// ---------------------------------------------------------------------------
__global__ __launch_bounds__(256)
void proj_gemm(const float* __restrict__ A, const float* __restrict__ W,
               float* __restrict__ Out, int ldw) {
  const int m0   = blockIdx.x * 16;
  const int n0   = threadIdx.y * 16;
  const int lane = threadIdx.x;          // 0..31 (wave32)
  const int row  = lane & 15;
  const int half = lane >> 4;            // 0: K={kk,kk+1}, 1: K={kk+2,kk+3}

  // A-fragment source: row (m0+row), starting col 2*half
  const float* a = A + (size_t)(m0 + row) * EMB + 2 * half;
  // B-fragment source: B[k][n] = W[n][k] -> row (n0+row) of W, col 2*half
  const float* w = W + (size_t)(n0 + row) * ldw + 2 * half;

  v8f acc = {};
#pragma unroll
  for (int kk = 0; kk < EMB; kk += 4) {
    v2f av = *(const v2f*)(a + kk);
    v2f bv = *(const v2f*)(w + kk);
    // (neg_a, A, neg_b, B, c_mod, C, reuse_a, reuse_b)
    acc = __builtin_amdgcn_wmma_f32_16x16x4_f32(
        false, av, false, bv, (short)0, acc, false, false);
  }

  // D layout: VGPR r -> row m0 + 8*half + r, col n0 + (lane&15)
  float* o = Out + (size_t)(m0 + 8 * half) * CNUM + n0 + row;
#pragma unroll
  for (int r = 0; r < 8; ++r) o[(size_t)r * CNUM] = acc[r];
}

// ---------------------------------------------------------------------------
// Kernel 2: edge phase. One wave per edge; each lane owns 4 channels.
// ---------------------------------------------------------------------------
__device__ __forceinline__ float sigf(float x) {
  return 1.0f / (1.0f + __expf(-x));     // v_exp_f32 path
}

__global__ __launch_bounds__(256)
void edge_kernel(const int* __restrict__ stu_idx, const int* __restrict__ item_idx,
                 const int* __restrict__ conc_idx,
                 const float* __restrict__ Pstu, const float* __restrict__ Pitem,
                 const float* __restrict__ Cstu, const float* __restrict__ Citem,
                 const float* __restrict__ w_pred, const float* __restrict__ b_pred,
                 float* __restrict__ out) {
  const int e    = blockIdx.x * 8 + threadIdx.y;   // 31250*8 == EDGES exactly
  const int lane = threadIdx.x;
  const int c0   = lane * 4;                       // 4 channels per lane

  const int s  = stu_idx[e];
  const int it = item_idx[e];

  const float4 sp = *(const float4*)(Pstu  + (size_t)s  * CNUM + c0);
  const float4 ip = *(const float4*)(Pitem + (size_t)it * CNUM + c0);
  const float4 wp = *(const float4*)(w_pred + c0);

  float acc[KCONC];
#pragma unroll
  for (int k = 0; k < KCONC; ++k) {
    const int ci = conc_idx[(size_t)e * KCONC + k];
    const float4 cs = *(const float4*)(Cstu  + (size_t)ci * CNUM + c0);
    const float4 ct = *(const float4*)(Citem + (size_t)ci * CNUM + c0);
    float sum;
    sum  = (sigf(sp.x + cs.x) - sigf(ip.x + ct.x)) * wp.x;
    sum += (sigf(sp.y + cs.y) - sigf(ip.y + ct.y)) * wp.y;
    sum += (sigf(sp.z + cs.z) - sigf(ip.z + ct.z)) * wp.z;
    sum += (sigf(sp.w + cs.w) - sigf(ip.w + ct.w)) * wp.w;
    acc[k] = sum;
  }

  // wave32 butterfly reduction per k
#pragma unroll
  for (int k = 0; k < KCONC; ++k) {
#pragma unroll
    for (int off = 16; off > 0; off >>= 1)
      acc[k] += __shfl_xor(acc[k], off, 32);
  }

  if (lane == 0) {
    const float b = b_pred[0];
    out[e] = 0.25f * (sigf(acc[0] + b) + sigf(acc[1] + b) +
                      sigf(acc[2] + b) + sigf(acc[3] + b));
  }
}

// ---------------------------------------------------------------------------
// Launcher
// ---------------------------------------------------------------------------
extern "C" void kernel_launch(void* const* d_in, const int* in_sizes, int n_in,
                              void* d_out, int out_size, void* d_ws, size_t ws_size,
                              hipStream_t stream) {
  const int*   stu_idx        = (const int*)d_in[0];
  const int*   item_idx       = (const int*)d_in[1];
  const int*   conc_idx       = (const int*)d_in[2];
  const float* stu_fusion     = (const float*)d_in[3];
  const float* item_fusion    = (const float*)d_in[4];
  const float* concept_fusion = (const float*)d_in[5];
  const float* W_stu          = (const float*)d_in[6];   // 128 x 128
  const float* W_item         = (const float*)d_in[7];   // 128 x 128
  const float* w_pred         = (const float*)d_in[8];   // 128
  const float* b_pred         = (const float*)d_in[9];   // 1
  float*       out            = (float*)d_out;

  // Workspace: P_stu | P_item | C_stu | C_item  (~32.8 MB total)
  float* Pstu  = (float*)d_ws;
  float* Pitem = Pstu  + (size_t)N_STU  * CNUM;
  float* Cstu  = Pitem + (size_t)N_ITEM * CNUM;
  float* Citem = Cstu  + (size_t)N_CONC * CNUM;

  const dim3 blk(32, 8);

  // stu_part  = stu_fusion  @ Ws_a^T  (Ws_a = W_stu[:, :64])
  proj_gemm<<<dim3(N_STU  / 16), blk, 0, stream>>>(stu_fusion,  W_stu,       Pstu,  2 * EMB);
  // item_part = item_fusion @ Wi_a^T  (Wi_a = W_item[:, :64])
  proj_gemm<<<dim3(N_ITEM / 16), blk, 0, stream>>>(item_fusion, W_item,      Pitem, 2 * EMB);
  // conc_stu  = concept_fusion @ Ws_b^T  (Ws_b = W_stu[:, 64:])
  proj_gemm<<<dim3(N_CONC / 16), blk, 0, stream>>>(concept_fusion, W_stu  + EMB, Cstu,  2 * EMB);
  // conc_item = concept_fusion @ Wi_b^T  (Wi_b = W_item[:, 64:])
  proj_gemm<<<dim3(N_CONC / 16), blk, 0, stream>>>(concept_fusion, W_item + EMB, Citem, 2 * EMB);

  // Edge phase (stream-ordered after the GEMMs)
  edge_kernel<<<dim3(EDGES / 8), blk, 0, stream>>>(
      stu_idx, item_idx, conc_idx, Pstu, Pitem, Cstu, Citem, w_pred, b_pred, out);
}